// EMAQuantizer_69664369541363
// MI455X (gfx1250) — compile-verified
//
#include <hip/hip_runtime.h>

// VQ-VAE EMA quantizer forward, MI455X (gfx1250, wave32).
// z:[16,64,64,64] NCHW fp32, embedding:[1024,64] fp32.
// Outputs (concat): quantized NCHW (4194304), loss (1), indices as float (65536), perplexity (1).

typedef __attribute__((ext_vector_type(2))) float v2f;
typedef __attribute__((ext_vector_type(8))) float v8f;

#define HW_        4096            // H*W
#define CCH_       64              // channels = D
#define K_         1024            // codebook size
#define D_         64
#define RSTRIDE    66              // padded LDS row stride (floats): banks (2k+d0)%64 distinct over 16 lanes
#define NROWS      65536           // B*H*W
#define ROWS_PER_BLOCK 128
#define NBLOCKS    (NROWS / ROWS_PER_BLOCK)   // 512
#define QELEMS     4194304         // N*D
// ws layout (float units): [0,1024) eNorm | [1024,2048) int hist | [2048,2560) loss partials

__global__ void vq_init(float* ws) {
    int t = blockIdx.x * 256 + threadIdx.x;
    if (t < 1536) ((int*)ws)[1024 + t] = 0;     // zero hist + loss partials
}

__global__ void vq_enorm(const float* __restrict__ emb, float* __restrict__ ws) {
    int k = blockIdx.x * 256 + threadIdx.x;
    if (k < K_) {
        const float* row = emb + k * D_;
        float s = 0.f;
        #pragma unroll
        for (int d = 0; d < D_; ++d) s = fmaf(row[d], row[d], s);
        ws[k] = s;
    }
}

__launch_bounds__(256, 1)
__global__ void vq_main(const float* __restrict__ z, const float* __restrict__ emb,
                        const float* __restrict__ eNormG, int* __restrict__ hist,
                        float* __restrict__ lossPart,
                        float* __restrict__ outQ, float* __restrict__ outIdx) {
    __shared__ float sEmb[K_ * RSTRIDE];   // 270336 B: full codebook resident in LDS
    __shared__ float sNorm[K_];            // 4 KB
    __shared__ int   sIdx[8][16];          // per-wave argmin results
    __shared__ float sRed[256];            // loss reduction

    const int tid = threadIdx.x;

    // Stage full codebook into padded LDS (float2 granularity keeps 8B alignment).
    for (int i = tid; i < K_ * 32; i += 256) {
        int row = i >> 5, c2 = (i & 31) << 1;
        v2f v = *(const v2f*)(emb + row * D_ + c2);
        *(v2f*)(&sEmb[row * RSTRIDE + c2]) = v;
    }
    for (int i = tid; i < K_; i += 256) sNorm[i] = eNormG[i];
    __syncthreads();

    const int wave = tid >> 5;
    const int lane = tid & 31;
    const int lm   = lane & 15;
    const int hi   = lane >> 4;                         // upper half-wave
    const int n    = blockIdx.x * ROWS_PER_BLOCK + wave * 16 + lm;
    const int b    = n >> 12;                           // n / (H*W)
    const int hw   = n & (HW_ - 1);
    const float* zbase = z + (size_t)b * (CCH_ * HW_) + hw;

    // A fragments: ISA 16x4 fp32 layout -> lane holds z[m=lane&15][d0], z[m][d0+1], d0 = 4t + 2*hi.
    v2f a[16];
    #pragma unroll
    for (int t = 0; t < 16; ++t) {
        int d0 = 4 * t + 2 * hi;
        a[t].x = zbase[(size_t)d0 * HW_];
        a[t].y = zbase[(size_t)(d0 + 1) * HW_];
    }

    float minS[8];
    int   minI[8];
    #pragma unroll
    for (int r = 0; r < 8; ++r) { minS[r] = 3.4e38f; minI[r] = 0x7fffffff; }

    // Sweep all 64 K-tiles: 16 x v_wmma_f32_16x16x4_f32 per tile, fused score + argmin.
    for (int kt = 0; kt < 64; ++kt) {
        const int code = kt * 16 + lm;
        const float* bbase = &sEmb[code * RSTRIDE + 2 * hi];  // B frag: e[code][d0], e[code][d0+1]
        v8f acc = {0.f, 0.f, 0.f, 0.f, 0.f, 0.f, 0.f, 0.f};
        #pragma unroll
        for (int t = 0; t < 16; ++t) {
            v2f bf = *(const v2f*)(bbase + 4 * t);
            acc = __builtin_amdgcn_wmma_f32_16x16x4_f32(
                false, a[t], false, bf, (short)0, acc, false, false);
        }
        float en = sNorm[code];
        #pragma unroll
        for (int r = 0; r < 8; ++r) {
            float s = fmaf(-2.0f, acc[r], en);   // ||e||^2 - 2 z.e (||z||^2 constant per row)
            if (s < minS[r]) { minS[r] = s; minI[r] = code; }  // strict < keeps lowest index on ties
        }
    }

    // Butterfly across each 16-lane half (rows 0-7 in lanes<16, rows 8-15 in lanes>=16).
    #pragma unroll
    for (int m = 1; m <= 8; m <<= 1) {
        #pragma unroll
        for (int r = 0; r < 8; ++r) {
            float os = __shfl_xor(minS[r], m, 32);
            int   oi = __shfl_xor(minI[r], m, 32);
            if (os < minS[r] || (os == minS[r] && oi < minI[r])) { minS[r] = os; minI[r] = oi; }
        }
    }
    if (lane == 0) {
        #pragma unroll
        for (int r = 0; r < 8; ++r) sIdx[wave][r] = minI[r];
    }
    if (lane == 16) {
        #pragma unroll
        for (int r = 0; r < 8; ++r) sIdx[wave][8 + r] = minI[r];
    }
    __syncthreads();

    const int idx = sIdx[wave][lm];
    if (hi == 0) {
        outIdx[n] = (float)idx;                 // indices output (as fp32 value)
        atomicAdd(&hist[idx], 1);               // integer atomic: deterministic
    }

    // Quantized output (straight-through forward value == embedding[idx]) + loss residual.
    float lacc = 0.f;
    float* obase = outQ + (size_t)b * (CCH_ * HW_) + hw;
    const float* erow = &sEmb[idx * RSTRIDE];
    #pragma unroll
    for (int t = 0; t < 16; ++t) {
        int d0 = 4 * t + 2 * hi;
        float e0 = erow[d0], e1 = erow[d0 + 1];
        obase[(size_t)d0 * HW_]       = e0;
        obase[(size_t)(d0 + 1) * HW_] = e1;
        float f0 = e0 - a[t].x, f1 = e1 - a[t].y;
        lacc = fmaf(f0, f0, lacc);
        lacc = fmaf(f1, f1, lacc);
    }

    // Deterministic fixed-order block reduction of loss partial.
    sRed[tid] = lacc;
    __syncthreads();
    for (int s = 128; s > 0; s >>= 1) {
        if (tid < s) sRed[tid] += sRed[tid + s];
        __syncthreads();
    }
    if (tid == 0) lossPart[blockIdx.x] = sRed[0];
}

__global__ void vq_final(const float* __restrict__ ws, float* __restrict__ d_out) {
    __shared__ float red[256];
    const int t = threadIdx.x;
    const int*   hist  = (const int*)ws + 1024;
    const float* parts = ws + 2048;

    // Loss: sum 512 per-block partials in fixed order.
    red[t] = parts[t] + parts[t + 256];
    __syncthreads();
    for (int s = 128; s > 0; s >>= 1) {
        if (t < s) red[t] += red[t + s];
        __syncthreads();
    }
    float loss = 0.25f * red[0] / 4194304.0f;
    __syncthreads();

    // Perplexity: exp(-sum p*log(p+1e-10)), p = count / N.
    float es = 0.f;
    for (int i = t; i < K_; i += 256) {
        float p = (float)hist[i] * (1.0f / 65536.0f);
        es += p * __logf(p + 1e-10f);
    }
    red[t] = es;
    __syncthreads();
    for (int s = 128; s > 0; s >>= 1) {
        if (t < s) red[t] += red[t + s];
        __syncthreads();
    }
    if (t == 0) {
        d_out[QELEMS]              = loss;            // loss scalar
        d_out[QELEMS + 1 + 65536]  = __expf(-red[0]); // perplexity scalar
    }
}

extern "C" void kernel_launch(void* const* d_in, const int* in_sizes, int n_in,
                              void* d_out, int out_size, void* d_ws, size_t ws_size,
                              hipStream_t stream) {
    const float* z   = (const float*)d_in[0];
    const float* emb = (const float*)d_in[1];
    float* ws  = (float*)d_ws;
    float* out = (float*)d_out;

    vq_init <<<6,       256, 0, stream>>>(ws);
    vq_enorm<<<4,       256, 0, stream>>>(emb, ws);
    vq_main <<<NBLOCKS, 256, 0, stream>>>(z, emb, ws, (int*)ws + 1024, ws + 2048,
                                          out, out + QELEMS + 1);
    vq_final<<<1,       256, 0, stream>>>(ws, out);
}